// STNtransformer_91302414778601
// MI455X (gfx1250) — compile-verified
//
#include <hip/hip_runtime.h>

// STN bilinear sampling, B=16, H=W=OH=OW=256, C=64, fp32.
// Memory-bound gather: B128-coalesced gathers + NT output stream.
// Block-uniform batch/row derivation so theta goes through the scalar path.

typedef float fvec4 __attribute__((ext_vector_type(4)));

__global__ __launch_bounds__(256) void stn_bilinear_kernel(
    const float* __restrict__ X,      // [B,H,W,C]
    const float* __restrict__ theta,  // [B,6]
    float* __restrict__ out)          // [B,OH,OW,C]
{
    constexpr int H = 256, W = 256, C = 64, CV = C / 4;  // CV = 16 float4/pixel

    // Block layout: 256 threads = 16 pixels x 16 float4-channels.
    // blockIdx.x: [ b (4b) | oy (8b) | ox-block (4b) ]  -> all wave-uniform.
    const int bx     = blockIdx.x;
    const int b      = bx >> 12;               // uniform: batch
    const int oy     = (bx >> 4) & (H - 1);    // uniform: output row
    const int oxbase = (bx & 15) << 4;         // uniform: first pixel of block
    const int sub    = threadIdx.x >> 4;       // 0..15: pixel within block
    const int vec    = threadIdx.x & (CV - 1); // 0..15: float4 within channels
    const int ox     = oxbase + sub;

    // linspace(-1,1,256): step = 2/255
    const float xc = __fmaf_rn((float)ox, 2.0f / 255.0f, -1.0f); // per 16-lane grp
    const float yc = __fmaf_rn((float)oy, 2.0f / 255.0f, -1.0f); // uniform

    // b is uniform -> these become s_load through the scalar cache.
    const float* th = theta + b * 6;
    const float t0 = th[0], t1 = th[1], t2 = th[2];
    const float t3 = th[3], t4 = th[4], t5 = th[5];

    // uniform halves fold to SALU; only the xc term is per-lane.
    const float sx = __fmaf_rn(t0, xc, __fmaf_rn(t1, yc, t2));
    const float sy = __fmaf_rn(t3, xc, __fmaf_rn(t4, yc, t5));

    // NOTE: reference scales by W (not W-1)
    const float x = 0.5f * (sx + 1.0f) * (float)W;
    const float y = 0.5f * (sy + 1.0f) * (float)H;

    // astype(int32) truncates toward zero, same as C cast
    const int xi = (int)x;
    const int yi = (int)y;
    const int x0i = min(max(xi,     0), W - 1);
    const int x1i = min(max(xi + 1, 0), W - 1);
    const int y0i = min(max(yi,     0), H - 1);
    const int y1i = min(max(yi + 1, 0), H - 1);

    const float dx1 = (float)x1i - x;
    const float dx0 = x - (float)x0i;
    const float dy1 = (float)y1i - y;
    const float dy0 = y - (float)y0i;
    const float wa = dx1 * dy1;
    const float wb = dx1 * dy0;
    const float wc = dx0 * dy1;
    const float wd = dx0 * dy0;

    // float4-granular addressing: ((b*H + y)*W + x)*CV + vec
    const fvec4* __restrict__ Xv = (const fvec4*)X;
    const size_t bbase = (size_t)b * ((size_t)H * W * CV);
    const size_t r0 = bbase + (size_t)y0i * (W * CV);
    const size_t r1 = bbase + (size_t)y1i * (W * CV);
    const size_t c0 = (size_t)x0i * CV + vec;
    const size_t c1 = (size_t)x1i * CV + vec;

    // 4 independent B128 gathers; 16 lanes -> one contiguous 256B burst each.
    const fvec4 pa = Xv[r0 + c0];
    const fvec4 pb = Xv[r1 + c0];
    const fvec4 pc = Xv[r0 + c1];
    const fvec4 pd = Xv[r1 + c1];

    fvec4 r = pa * wa;
    r = pb * wb + r;
    r = pc * wc + r;
    r = pd * wd + r;

    // Store index == global tid -> fully contiguous stream. Non-temporal so the
    // 256MB write-once stream doesn't evict X (256MB vs 192MB global L2): X's
    // 4-neighbor gather reuse in L2 decides effective HBM read traffic.
    const int tid = bx * 256 + threadIdx.x;
    __builtin_nontemporal_store(r, ((fvec4*)out) + tid);
}

extern "C" void kernel_launch(void* const* d_in, const int* in_sizes, int n_in,
                              void* d_out, int out_size, void* d_ws, size_t ws_size,
                              hipStream_t stream) {
    const float* X     = (const float*)d_in[0];   // [16,256,256,64] fp32
    const float* theta = (const float*)d_in[1];   // [16,6] fp32
    float* out         = (float*)d_out;           // [16,256,256,64] fp32

    constexpr int B = 16, OH = 256, OW = 256, CV = 16;
    const int total_threads = B * OH * OW * CV;   // 16,777,216
    const int block = 256;                        // 8 waves (wave32)
    const int grid  = total_threads / block;      // 65,536 blocks

    hipLaunchKernelGGL(stn_bilinear_kernel, dim3(grid), dim3(block), 0, stream,
                       X, theta, out);
}